// EdgeConv_81965155877667
// MI455X (gfx1250) — compile-verified
//
#include <hip/hip_runtime.h>

#define B_  8
#define C_  64
#define N_  4096
#define O_  64
#define K_  16
#define C2_ 128
#define NCHUNK_ (N_ / 64)
#define BT_STRIDE_ 66                  // padded LDS row stride (floats), via TDM pad
#define CNT_ ((float)(B_ * N_ * K_))   // BN reduction count per channel
#define BN_EPS 1e-5f
#define LRELU 0.2f
#define NEG_INF (-3.402823466e38f)

typedef __attribute__((ext_vector_type(2))) float v2f;
typedef __attribute__((ext_vector_type(4))) float v4f;
typedef __attribute__((ext_vector_type(8))) float v8f;
typedef __attribute__((ext_vector_type(4))) unsigned int u32x4;
typedef __attribute__((ext_vector_type(8))) int i32x8;
typedef __attribute__((ext_vector_type(4))) int i32x4;

__device__ __forceinline__ v8f wmma4(v2f a, v2f b, v8f c) {
  // D = A(16x4,f32) * B(4x16,f32) + C(16x16,f32)
  return __builtin_amdgcn_wmma_f32_16x16x4_f32(false, a, false, b, (short)0, c, false, false);
}

// ---------------------------------------------------------------------------
// TDM: one-row 1-D tile of `n` f32 elements, global -> LDS. With pad!=0, LDS
// destination gets 2 dwords of padding after every 64 dwords (row stride 66),
// which de-conflicts the 64-bank LDS for the WMMA fragment reads.
// Descriptor per CDNA5 ISA §8.3/8.4 (D# group0/group1); groups 2/3 unused.
// Toolchain uses the 6-arg builtin: (u32x4, i32x8, i32x4, i32x4, i32x8, cpol).
// ---------------------------------------------------------------------------
__device__ __forceinline__ void tdm_load_f32(void* lds, const void* g, int n, int pad) {
  unsigned long long ga = (unsigned long long)(size_t)g;
  u32x4 g0;
  g0[0] = 1u;                                                   // count=1, user D#
  g0[1] = (unsigned int)(size_t)lds;                            // LDS byte addr (addr[31:0])
  g0[2] = (unsigned int)ga;                                     // global_addr[31:0]
  g0[3] = (unsigned int)((ga >> 32) & 0x01FFFFFFu) | (2u << 30); // addr[56:32] | type=2
  i32x8 g1;
  g1[0] = (2 << 16)                                             // data_size = 4B
        | (pad ? ((1 << 20) | (5 << 22) | (1 << 25)) : 0);      // pad_en, 64dw interval, 2dw pad
  g1[1] = (n & 0xFFFF) << 16;                                   // tensor_dim0[15:0]
  g1[2] = ((n >> 16) & 0xFFFF) | (1 << 16);                     // tensor_dim0[31:16] | tensor_dim1=1
  g1[3] = (n & 0xFFFF) << 16;                                   // tile_dim0
  g1[4] = 1;                                                    // tile_dim1=1, tile_dim2=0
  g1[5] = n;                                                    // tensor_dim0_stride[31:0]
  g1[6] = 0;
  g1[7] = 0;
  i32x4 z4 = {0, 0, 0, 0};
  i32x8 z8 = {0, 0, 0, 0, 0, 0, 0, 0};
  __builtin_amdgcn_tensor_load_to_lds(g0, g1, z4, z4, z8, 0);
}

// ---------------------------------------------------------------------------
// K1: transpose x (B,C,N) -> xt (B,N,C) and per-point squared norms xx (B,N)
// ---------------------------------------------------------------------------
__global__ __launch_bounds__(256) void xt_norm_kernel(const float* __restrict__ x,
                                                      float* __restrict__ xt,
                                                      float* __restrict__ xx) {
  int gid = blockIdx.x * blockDim.x + threadIdx.x;   // over B*N
  if (gid >= B_ * N_) return;
  int b = gid / N_, n = gid % N_;
  const float* xp = x + (size_t)b * C_ * N_ + n;
  float* xtp = xt + (size_t)gid * C_;
  float s = 0.f;
#pragma unroll
  for (int c = 0; c < C_; ++c) {
    float v = xp[(size_t)c * N_];
    xtp[c] = v;
    s += v * v;
  }
  xx[gid] = s;
}

// ---------------------------------------------------------------------------
// K2: fused distance-GEMM + top-16.
//  - B tiles double-buffered in LDS via TDM (tensor_load_to_lds + tensorcnt)
//  - 16 batched fragment loads, then 16 back-to-back WMMAs
//  - parallel threshold prefilter into per-row LDS queues (ds atomics),
//    serial insertion only on queue survivors
// ---------------------------------------------------------------------------
__global__ __launch_bounds__(128) void knn_kernel(const float* __restrict__ xt,
                                                  const float* __restrict__ xx,
                                                  int* __restrict__ idxout) {
  const int tile    = blockIdx.x;            // B * (N/16) tiles
  const int b       = tile / (N_ / 16);
  const int rowBase = (tile % (N_ / 16)) * 16;
  const float* xtb  = xt + (size_t)b * N_ * C_;
  const float* xxb  = xx + (size_t)b * N_;

  const int tid  = threadIdx.x;
  const int lane = tid & 31;
  const int wave = tid >> 5;
  const int m    = lane & 15;
  const int koff = (lane >> 4) * 2;

  __shared__ float btile[2][64 * BT_STRIDE_];  // 2 x 16.5KB, padded rows
  __shared__ float lds_dist[16][64];
  __shared__ float row_thresh[16];
  __shared__ int   qcount[16];
  __shared__ int   qlist[16][64];
  __shared__ float bestv[16][17];
  __shared__ int   besti[16][17];

  if (tid < 16) { row_thresh[tid] = NEG_INF; qcount[tid] = 0; }
#pragma unroll
  for (int i = tid; i < 256; i += 128) {
    bestv[i >> 4][i & 15] = NEG_INF;
    besti[i >> 4][i & 15] = 0;
  }

  // Preload A fragments (row block fixed per WG): batched v2f global loads
  v2f areg[16];
  {
    const float* ap = xtb + (size_t)(rowBase + m) * C_;
#pragma unroll
    for (int s = 0; s < 16; ++s) areg[s] = *(const v2f*)(ap + 4 * s + koff);
  }

  // Kick TDM for chunk 0 and wait (wave 0 only; barrier publishes to all)
  if (wave == 0) {
    tdm_load_f32(&btile[0][0], xtb, 64 * C_, 1);
    __builtin_amdgcn_s_wait_tensorcnt(0);
  }
  __syncthreads();

  for (int c = 0; c < NCHUNK_; ++c) {
    // ---- compute 16x16 score tile per wave from LDS B tile ----
    const float* bp = &btile[c & 1][(wave * 16 + m) * BT_STRIDE_];
    v2f bfrag[16];
#pragma unroll
    for (int s = 0; s < 16; ++s) bfrag[s] = *(const v2f*)(bp + 4 * s + koff);
    v8f acc = {};
#pragma unroll
    for (int s = 0; s < 16; ++s) acc = wmma4(areg[s], bfrag[s], acc);

    // ---- prefetch next chunk's B tile into the other LDS buffer ----
    if (wave == 0 && (c + 1) < NCHUNK_) {
      tdm_load_f32(&btile[(c + 1) & 1][0], xtb + (size_t)(c + 1) * 64 * C_, 64 * C_, 1);
    }

    const float xxn = xxb[c * 64 + wave * 16 + m];
#pragma unroll
    for (int j = 0; j < 8; ++j) {
      int row = (lane < 16) ? j : (j + 8);
      lds_dist[row][wave * 16 + m] = 2.0f * acc[j] - xxn;   // row-const -xx[m] dropped
    }
    __syncthreads();   // dist visible

    // ---- parallel prefilter: 8 threads/row, 8 cols each ----
    {
      const int row = tid >> 3;
      const int c0  = (tid & 7) * 8;
      const float th = row_thresh[row];
#pragma unroll
      for (int i = 0; i < 8; ++i) {
        float d = lds_dist[row][c0 + i];
        if (d > th) {
          int pos = atomicAdd(&qcount[row], 1);
          qlist[row][pos] = c0 + i;
        }
      }
    }
    __syncthreads();   // queues complete

    // ---- owners (threads 0..15, all in wave 0) consume queues ----
    if (tid < 16) {
      const int row = tid;
      const int qn  = qcount[row];
#pragma unroll 1
      for (int q = 0; q < qn; ++q) {
        int cc  = qlist[row][q];
        float d = lds_dist[row][cc];
        if (d > bestv[row][0]) {
          int pos = 0;
#pragma unroll 1
          while (pos < 15 && bestv[row][pos + 1] < d) {
            bestv[row][pos] = bestv[row][pos + 1];
            besti[row][pos] = besti[row][pos + 1];
            ++pos;
          }
          bestv[row][pos] = d;
          besti[row][pos] = c * 64 + cc;
        }
      }
      qcount[row] = 0;
      row_thresh[row] = bestv[row][0];
    }
    if (wave == 0) __builtin_amdgcn_s_wait_tensorcnt(0);  // next tile landed
    __syncthreads();
  }

  if (tid < 16) {
    int* op = idxout + ((size_t)b * N_ + rowBase + tid) * K_;
#pragma unroll
    for (int i = 0; i < 16; ++i) op[i] = besti[tid][15 - i];  // descending order
  }
}

// ---------------------------------------------------------------------------
// Shared tile core for K3/K5: one workgroup per (b,n).
// Rows = k (16), cols = channel o (16 per wave), K-dim = 128 (32 WMMA steps).
// W fragments preloaded to registers; feat staged in padded LDS (stride 132).
// ---------------------------------------------------------------------------
__device__ __forceinline__ v8f edge_tile(const float* __restrict__ xt,
                                         const int* __restrict__ idxbuf,
                                         const float* __restrict__ Wm,
                                         int b, int n,
                                         float (*feat)[132]) {
  const int tid  = threadIdx.x;
  const int lane = tid & 31;
  const int wave = tid >> 5;
  const int m    = lane & 15;
  const int koff = (lane >> 4) * 2;

  // Stage feat tile: 16 rows (k) x 128 (c); 8 threads/row, 16 c's each, b128 ops
  {
    const int row  = tid >> 3;
    const int part = (tid & 7) * 16;          // whole range < 64 or >= 64
    const int nb   = idxbuf[((size_t)b * N_ + n) * K_ + row];
    const float* xc = xt + ((size_t)b * N_ + n) * C_;
    const float* xn = xt + ((size_t)b * N_ + nb) * C_;
    if (part < C_) {
#pragma unroll
      for (int i = 0; i < 4; ++i) {
        v4f a = *(const v4f*)(xn + part + 4 * i);
        v4f c = *(const v4f*)(xc + part + 4 * i);
        *(v4f*)&feat[row][part + 4 * i] = a - c;
      }
    } else {
#pragma unroll
      for (int i = 0; i < 4; ++i)
        *(v4f*)&feat[row][part + 4 * i] = *(const v4f*)(xc + part - C_ + 4 * i);
    }
  }

  // Preload all 32 W fragments (overlaps with staging, before the barrier)
  const int o = wave * 16 + m;
  const float* wrow = Wm + (size_t)o * C2_;
  v2f wfrag[32];
#pragma unroll
  for (int s = 0; s < 32; ++s) wfrag[s] = *(const v2f*)(wrow + 4 * s + koff);

  __syncthreads();

  v8f acc = {};
  v2f afrag[16];
#pragma unroll
  for (int s = 0; s < 16; ++s) afrag[s] = *(const v2f*)(&feat[m][4 * s + koff]);
#pragma unroll
  for (int s = 0; s < 16; ++s) acc = wmma4(afrag[s], wfrag[s], acc);
#pragma unroll
  for (int s = 0; s < 16; ++s) afrag[s] = *(const v2f*)(&feat[m][64 + 4 * s + koff]);
#pragma unroll
  for (int s = 0; s < 16; ++s) acc = wmma4(afrag[s], wfrag[16 + s], acc);
  return acc;
}

// K3: accumulate per-channel sum / sumsq of y (global BN stats)
__global__ __launch_bounds__(128) void edge_stats_kernel(const float* __restrict__ xt,
                                                         const int* __restrict__ idxbuf,
                                                         const float* __restrict__ Wm,
                                                         float* __restrict__ stats) {
  __shared__ float feat[16][132];
  const int bn = blockIdx.x;
  const int b = bn / N_, n = bn % N_;
  v8f acc = edge_tile(xt, idxbuf, Wm, b, n, feat);

  const int lane = threadIdx.x & 31;
  const int o = (threadIdx.x >> 5) * 16 + (lane & 15);

  float s1 = 0.f, s2 = 0.f;
#pragma unroll
  for (int j = 0; j < 8; ++j) { float v = acc[j]; s1 += v; s2 += v * v; }
  s1 += __shfl_xor(s1, 16, 32);         // fold the two half-waves sharing o
  s2 += __shfl_xor(s2, 16, 32);
  if (lane < 16) {
    atomicAdd(&stats[o],      s1);
    atomicAdd(&stats[64 + o], s2);
  }
}

// K4: finalize BN stats
__global__ void finalize_stats_kernel(float* __restrict__ stats) {
  int t = threadIdx.x;
  if (t < O_) {
    float mean = stats[t] / CNT_;
    float var  = stats[64 + t] / CNT_ - mean * mean;
    stats[128 + t] = mean;
    stats[192 + t] = rsqrtf(var + BN_EPS);
  }
}

__global__ void zero_stats_kernel(float* __restrict__ stats) {
  int t = threadIdx.x;
  if (t < 256) stats[t] = 0.f;
}

// K5: recompute y tile, BN + LeakyReLU + max over k, write out (B,O,N)
__global__ __launch_bounds__(128) void edge_out_kernel(const float* __restrict__ xt,
                                                       const int* __restrict__ idxbuf,
                                                       const float* __restrict__ Wm,
                                                       const float* __restrict__ stats,
                                                       const float* __restrict__ gamma,
                                                       const float* __restrict__ beta,
                                                       float* __restrict__ out) {
  __shared__ float feat[16][132];
  const int bn = blockIdx.x;
  const int b = bn / N_, n = bn % N_;
  v8f acc = edge_tile(xt, idxbuf, Wm, b, n, feat);

  const int lane = threadIdx.x & 31;
  const int o = (threadIdx.x >> 5) * 16 + (lane & 15);

  const float mean = stats[128 + o];
  const float istd = stats[192 + o];
  const float g = gamma[o], bt = beta[o];

  float mx = NEG_INF;
#pragma unroll
  for (int j = 0; j < 8; ++j) {
    float y = (acc[j] - mean) * istd * g + bt;
    y = (y >= 0.f) ? y : LRELU * y;
    mx = fmaxf(mx, y);
  }
  mx = fmaxf(mx, __shfl_xor(mx, 16, 32));  // combine k=0..7 with k=8..15
  if (lane < 16) out[((size_t)b * O_ + o) * N_ + n] = mx;
}

// ---------------------------------------------------------------------------
extern "C" void kernel_launch(void* const* d_in, const int* in_sizes, int n_in,
                              void* d_out, int out_size, void* d_ws, size_t ws_size,
                              hipStream_t stream) {
  (void)in_sizes; (void)n_in; (void)out_size; (void)ws_size;
  const float* x     = (const float*)d_in[0];
  const float* Wm    = (const float*)d_in[1];
  const float* gamma = (const float*)d_in[2];
  const float* beta  = (const float*)d_in[3];
  float* out = (float*)d_out;

  // Workspace layout
  float* xt    = (float*)d_ws;                          // B*N*C = 2,097,152 f
  float* xx    = xt + (size_t)B_ * N_ * C_;             // B*N   =    32,768 f
  int*   idx   = (int*)(xx + (size_t)B_ * N_);          // B*N*K =   524,288 i
  float* stats = (float*)(idx + (size_t)B_ * N_ * K_);  // 256 f: sum|sumsq|mean|invstd

  hipLaunchKernelGGL(xt_norm_kernel, dim3((B_ * N_ + 255) / 256), dim3(256), 0, stream,
                     x, xt, xx);
  hipLaunchKernelGGL(zero_stats_kernel, dim3(1), dim3(256), 0, stream, stats);
  hipLaunchKernelGGL(knn_kernel, dim3(B_ * (N_ / 16)), dim3(128), 0, stream,
                     xt, xx, idx);
  hipLaunchKernelGGL(edge_stats_kernel, dim3(B_ * N_), dim3(128), 0, stream,
                     xt, idx, Wm, stats);
  hipLaunchKernelGGL(finalize_stats_kernel, dim3(1), dim3(64), 0, stream, stats);
  hipLaunchKernelGGL(edge_out_kernel, dim3(B_ * N_), dim3(128), 0, stream,
                     xt, idx, Wm, stats, gamma, beta, out);
}